// KnowledgeGraphEncoder_67044439491163
// MI455X (gfx1250) — compile-verified
//
#include <hip/hip_runtime.h>

#define N_NODES 100000
#define N_EDGES 1600000
#define CH 64
#define BN_EPS 1e-5f

typedef __attribute__((ext_vector_type(2))) float v2f;
typedef __attribute__((ext_vector_type(8))) float v8f;

// ---------------------------------------------------------------------------
// Degree / normalization kernels
// ---------------------------------------------------------------------------
__global__ __launch_bounds__(256) void k_init_deg(float* __restrict__ deg) {
    int i = blockIdx.x * 256 + threadIdx.x;
    if (i < N_NODES) deg[i] = 1.0f;  // self-loop
}

__global__ __launch_bounds__(256) void k_edge_deg(const int* __restrict__ dst,
                                                  float* __restrict__ deg) {
    int e = blockIdx.x * 256 + threadIdx.x;
    if (e < N_EDGES) atomicAdd(&deg[dst[e]], 1.0f);
}

__global__ __launch_bounds__(256) void k_to_dis(float* __restrict__ deg) {
    int i = blockIdx.x * 256 + threadIdx.x;
    if (i < N_NODES) deg[i] = rsqrtf(deg[i]);
}

// ---------------------------------------------------------------------------
// Dense GEMM: H[nrows,64] = X[nrows,64] @ W[64,64], f32 WMMA 16x16x4.
// Block = 256 threads (8 waves), 128 rows per block. LDS staged, stride 68
// (keeps 16B alignment for b128 LDS stores, avoids bank conflicts on the
// column reads of the A fragments).
// ---------------------------------------------------------------------------
#define XS 68
#define WS2 68

__global__ __launch_bounds__(256) void k_gemm_wmma(const float* __restrict__ X,
                                                   const float* __restrict__ W,
                                                   float* __restrict__ H,
                                                   int nrows) {
    __shared__ float sX[128 * XS];
    __shared__ float sW[64 * WS2];

    const int tid   = threadIdx.x;
    const int mbase = blockIdx.x * 128;

    // Stage W (64x64) coalesced: 1024 float4
    for (int i = tid; i < 64 * 16; i += 256) {
        int r  = i >> 4;
        int c4 = (i & 15) << 2;
        float4 w4 = ((const float4*)W)[i];
        *(float4*)&sW[r * WS2 + c4] = w4;
    }
    // Stage X rows [mbase, mbase+128), zero-padded past nrows: 2048 float4
    for (int i = tid; i < 128 * 16; i += 256) {
        int r    = i >> 4;
        int c4   = (i & 15) << 2;
        int grow = mbase + r;
        float4 xv = make_float4(0.f, 0.f, 0.f, 0.f);
        if (grow < nrows) xv = ((const float4*)(X + (size_t)grow * CH))[c4 >> 2];
        *(float4*)&sX[r * XS + c4] = xv;
    }
    __syncthreads();

    const int wave  = tid >> 5;
    const int lane  = tid & 31;
    const int mrow  = lane & 15;
    const int khalf = (lane >> 4) << 1;  // 0 or 2 (per ISA 32-bit A/B layout)
    const int ncol  = lane & 15;

    const float* aRow = &sX[(wave * 16 + mrow) * XS];

    v8f acc0 = {}, acc1 = {}, acc2 = {}, acc3 = {};

    for (int kb = 0; kb < 16; ++kb) {
        const int k4 = kb * 4;
        v2f a;
        a.x = aRow[k4 + khalf];
        a.y = aRow[k4 + khalf + 1];
        const float* bR0 = &sW[(k4 + khalf) * WS2];
        const float* bR1 = &sW[(k4 + khalf + 1) * WS2];
        v2f b0; b0.x = bR0[ 0 + ncol]; b0.y = bR1[ 0 + ncol];
        v2f b1; b1.x = bR0[16 + ncol]; b1.y = bR1[16 + ncol];
        v2f b2; b2.x = bR0[32 + ncol]; b2.y = bR1[32 + ncol];
        v2f b3; b3.x = bR0[48 + ncol]; b3.y = bR1[48 + ncol];
        acc0 = __builtin_amdgcn_wmma_f32_16x16x4_f32(false, a, false, b0, (short)0, acc0, false, false);
        acc1 = __builtin_amdgcn_wmma_f32_16x16x4_f32(false, a, false, b1, (short)0, acc1, false, false);
        acc2 = __builtin_amdgcn_wmma_f32_16x16x4_f32(false, a, false, b2, (short)0, acc2, false, false);
        acc3 = __builtin_amdgcn_wmma_f32_16x16x4_f32(false, a, false, b3, (short)0, acc3, false, false);
    }

    // D layout: VGPR r -> M = r (lanes 0-15) / r+8 (lanes 16-31); N = lane&15
    const int mhi = (lane >> 4) << 3;
    for (int r = 0; r < 8; ++r) {
        int grow = mbase + wave * 16 + r + mhi;
        if (grow < nrows) {
            float* o = H + (size_t)grow * CH;
            o[ 0 + ncol] = acc0[r];
            o[16 + ncol] = acc1[r];
            o[32 + ncol] = acc2[r];
            o[48 + ncol] = acc3[r];
        }
    }
}

// ---------------------------------------------------------------------------
// Edge scatter: 64 threads per edge, one channel each (coalesced gather +
// coalesced float atomics). src/dst uniform per wave -> scalarize.
// ---------------------------------------------------------------------------
__global__ __launch_bounds__(256) void k_scatter(const float* __restrict__ H,
                                                 const int* __restrict__ src,
                                                 const int* __restrict__ dst,
                                                 const float* __restrict__ dis,
                                                 float* __restrict__ agg) {
    unsigned tid = blockIdx.x * 256u + threadIdx.x;
    unsigned e   = tid >> 6;
    if (e >= N_EDGES) return;
    int c = tid & 63;
    int s = __builtin_amdgcn_readfirstlane(src[e]);
    int d = __builtin_amdgcn_readfirstlane(dst[e]);
    float coef = dis[s] * dis[d];
    atomicAdd(&agg[(size_t)d * CH + c], H[(size_t)s * CH + c] * coef);
}

// ---------------------------------------------------------------------------
// Finalize: acc[i][c] += h[i][c]*dis[i]^2 + b[c]  (in place).
// Optionally accumulates BatchNorm sum / sum-of-squares per channel.
// Block = 64 nodes x 64 channels, each thread handles 16 elements.
// ---------------------------------------------------------------------------
template <bool STATS>
__global__ __launch_bounds__(256) void k_finalize(float* __restrict__ acc,
                                                  const float* __restrict__ H,
                                                  const float* __restrict__ dis,
                                                  const float* __restrict__ bias,
                                                  float* __restrict__ sumBuf) {
    __shared__ float sred[256];
    __shared__ float sred2[256];
    const int c = threadIdx.x & 63;
    const int g = threadIdx.x >> 6;  // 0..3
    const int nodeBase = blockIdx.x * 64;
    const float b = bias[c];
    float lsum = 0.f, lsq = 0.f;
    for (int j = 0; j < 16; ++j) {
        int i = nodeBase + g + j * 4;
        if (i < N_NODES) {
            float di = dis[i];
            size_t idx = (size_t)i * CH + c;
            float v = acc[idx] + H[idx] * di * di + b;
            acc[idx] = v;
            if (STATS) { lsum += v; lsq += v * v; }
        }
    }
    if (STATS) {
        sred[threadIdx.x] = lsum;
        sred2[threadIdx.x] = lsq;
        __syncthreads();
        if (threadIdx.x < 128) {
            sred[threadIdx.x]  += sred[threadIdx.x + 128];
            sred2[threadIdx.x] += sred2[threadIdx.x + 128];
        }
        __syncthreads();
        if (threadIdx.x < 64) {
            atomicAdd(&sumBuf[threadIdx.x],      sred[threadIdx.x]  + sred[threadIdx.x + 64]);
            atomicAdd(&sumBuf[64 + threadIdx.x], sred2[threadIdx.x] + sred2[threadIdx.x + 64]);
        }
    }
}

__global__ void k_bn_params(const float* __restrict__ sumBuf,
                            const float* __restrict__ gamma,
                            const float* __restrict__ beta,
                            float* __restrict__ ss) {
    int c = threadIdx.x;  // 64 threads
    const float invN = 1.0f / (float)N_NODES;
    float mu  = sumBuf[c] * invN;
    float var = sumBuf[64 + c] * invN - mu * mu;
    float sc  = gamma[c] * rsqrtf(var + BN_EPS);
    ss[c]      = sc;
    ss[64 + c] = beta[c] - mu * sc;
}

__global__ __launch_bounds__(256) void k_bn_relu(float* __restrict__ acc,
                                                 const float* __restrict__ ss) {
    size_t idx = (size_t)blockIdx.x * 256 + threadIdx.x;
    if (idx < (size_t)N_NODES * CH) {
        int c = idx & 63;
        float v = acc[idx] * ss[c] + ss[64 + c];
        acc[idx] = fmaxf(v, 0.0f);
    }
}

// ---------------------------------------------------------------------------
// Launcher
// ---------------------------------------------------------------------------
extern "C" void kernel_launch(void* const* d_in, const int* in_sizes, int n_in,
                              void* d_out, int out_size, void* d_ws, size_t ws_size,
                              hipStream_t stream) {
    const float* x     = (const float*)d_in[0];
    const int*   src   = (const int*)d_in[1];
    const int*   dst   = src + N_EDGES;
    const float* W1    = (const float*)d_in[2];
    const float* b1    = (const float*)d_in[3];
    const float* gamma = (const float*)d_in[4];
    const float* beta  = (const float*)d_in[5];
    const float* W2    = (const float*)d_in[6];
    const float* b2    = (const float*)d_in[7];
    float* out = (float*)d_out;

    // Workspace layout (floats)
    float* ws    = (float*)d_ws;
    float* dis   = ws;                   // N_NODES (padded to 100032)
    float* stats = ws + 100032;          // 128: sum[64], sumsq[64]
    float* ss    = ws + 100160;          // 128: scale[64], shift[64]
    float* h     = ws + 100288;          // N*64
    float* agg   = h + (size_t)N_NODES * CH;

    const int gridN   = (N_NODES + 255) / 256;          // 391
    const int gridE   = (N_EDGES + 255) / 256;          // 6250
    const int gridG   = (N_NODES + 127) / 128;          // 782 GEMM blocks
    const int gridS   = (int)(((size_t)N_EDGES * 64 + 255) / 256);  // 400000
    const int gridF   = (N_NODES + 63) / 64;            // 1563
    const int gridNC  = (int)(((size_t)N_NODES * CH + 255) / 256);  // 25000

    // deg -> dis
    k_init_deg<<<gridN, 256, 0, stream>>>(dis);
    k_edge_deg<<<gridE, 256, 0, stream>>>(dst, dis);
    k_to_dis<<<gridN, 256, 0, stream>>>(dis);

    // Layer 1: h = x @ W1; agg = scatter; agg = agg + h*dis^2 + b1 (+BN stats)
    k_gemm_wmma<<<gridG, 256, 0, stream>>>(x, W1, h, N_NODES);
    hipMemsetAsync(agg, 0, (size_t)N_NODES * CH * sizeof(float), stream);
    k_scatter<<<gridS, 256, 0, stream>>>(h, src, dst, dis, agg);
    hipMemsetAsync(stats, 0, 128 * sizeof(float), stream);
    k_finalize<true><<<gridF, 256, 0, stream>>>(agg, h, dis, b1, stats);

    // BatchNorm + ReLU (in place on agg)
    k_bn_params<<<1, 64, 0, stream>>>(stats, gamma, beta, ss);
    k_bn_relu<<<gridNC, 256, 0, stream>>>(agg, ss);

    // Layer 2: h = agg @ W2; scatter directly into d_out; finalize in d_out
    k_gemm_wmma<<<gridG, 256, 0, stream>>>(agg, W2, h, N_NODES);
    hipMemsetAsync(out, 0, (size_t)N_NODES * CH * sizeof(float), stream);
    k_scatter<<<gridS, 256, 0, stream>>>(h, src, dst, dis, out);
    k_finalize<false><<<gridF, 256, 0, stream>>>(out, h, dis, b2, nullptr);
}